// VariableSizedEmbedding_50148038148546
// MI455X (gfx1250) — compile-verified
//
#include <hip/hip_runtime.h>

typedef __attribute__((ext_vector_type(2))) float v2f;
typedef __attribute__((ext_vector_type(8))) float v8f;

#define NTOK   81920      // 4096 * 20 tokens
#define BCOUNT 25000      // entities per bucket
#define NENT   100000     // sentinel index
#define HDIM   64
#define DDIM   256

struct Params {
    const int*   ids;
    const float* emb[4];
    const float* w1[4];
    const float* b1[4];
    const float* w2[4];
    const float* b2[4];
    float*       out;
    const int*   counts;
    const int*   perm;
};

__global__ void zero_counts_kernel(int* counts) {
    if (threadIdx.x < 4) counts[threadIdx.x] = 0;
}

__global__ __launch_bounds__(256) void bucketize_kernel(const int* __restrict__ ids,
                                                        int* __restrict__ counts,
                                                        int* __restrict__ perm) {
    int tok = blockIdx.x * 256 + threadIdx.x;
    if (tok >= NTOK) return;
    int id = ids[tok];
    if (id == NENT) id = 0;              // sentinel -> padding row of bucket 0
    int b = id / BCOUNT;                 // buckets are contiguous, pre-sorted
    int pos = atomicAdd(&counts[b], 1);
    perm[b * NTOK + pos] = tok;
}

// One block = 16 tokens of ONE bucket. 128 threads = 4 wave32s.
// Layer1: [16 x s] @ [s x 64] + b1, ReLU   (wave w -> hidden cols 16w..16w+15)
// Layer2: [16 x 64] @ [64 x 256] + b2      (wave w -> out cols 64w..64w+63, 4 tiles)
__global__ __launch_bounds__(128) void mlp_gemm_kernel(Params p) {
    __shared__ int   tok_s[16];
    __shared__ int   loc_s[16];
    __shared__ float etile[16 * 132];   // 16 x s, stride s+4 (conflict-free)
    __shared__ float hbuf[16 * 68];     // 16 x 64, stride 68 (conflict-free)

    const int b       = blockIdx.y;
    const int s       = 16 << b;        // 16,32,64,128
    const int lg      = 4 + b;
    const int estride = s + 4;
    const int count   = p.counts[b];
    const int row0    = blockIdx.x << 4;
    if (row0 >= count) return;          // uniform exit: EXEC stays all-1s below

    const int tid  = threadIdx.x;
    const int lane = tid & 31;
    const int wave = tid >> 5;
    const int lo   = lane & 15;         // N / M index within fragment
    const int hi   = lane >> 4;         // half-wave select
    const int khi  = hi << 1;           // K offset contributed by half-wave

    const float* __restrict__ embb = p.emb[b];
    const float* __restrict__ w1b  = p.w1[b];
    const float* __restrict__ b1b  = p.b1[b];
    const float* __restrict__ w2b  = p.w2[b];
    const float* __restrict__ b2b  = p.b2[b];

    if (tid < 16) {
        int r = tid, t = -1, loc = 0;
        if (row0 + r < count) {
            t = p.perm[b * NTOK + row0 + r];
            int id = p.ids[t];
            if (id == NENT) id = 0;
            loc = id - b * BCOUNT;      // local row in this bucket's table
        }
        tok_s[r] = t;
        loc_s[r] = loc;
    }
    __syncthreads();

    // cooperative gather of the 16 x s embedding tile into LDS
    for (int i = tid; i < (s << 4); i += 128) {
        int r = i >> lg;
        int c = i & (s - 1);
        float v = 0.0f;
        if (tok_s[r] >= 0) v = embb[loc_s[r] * s + c];
        etile[r * estride + c] = v;
    }
    __syncthreads();

    // ---------------- layer 1 ----------------
    {
        const int colN = (wave << 4) + lo;          // hidden column 0..63
        v8f acc = {};
        for (int kk = 0; kk < s; kk += 4) {
            v2f a, bb;
            a.x  = etile[lo * estride + kk + khi];
            a.y  = etile[lo * estride + kk + khi + 1];
            bb.x = w1b[(kk + khi)     * HDIM + colN];
            bb.y = w1b[(kk + khi + 1) * HDIM + colN];
            acc = __builtin_amdgcn_wmma_f32_16x16x4_f32(
                false, a, false, bb, (short)0, acc, false, false);
        }
        const float bias = b1b[colN];
#pragma unroll
        for (int v = 0; v < 8; ++v) {
            float x = acc[v] + bias;                // M = v + 8*hi
            hbuf[(v + (hi << 3)) * 68 + colN] = fmaxf(x, 0.0f);
        }
    }
    __syncthreads();

    // ---------------- layer 2 ----------------
    for (int j = 0; j < 4; ++j) {
        const int col = (wave << 6) + (j << 4) + lo;    // output column 0..255
        v8f acc = {};
        for (int kk = 0; kk < HDIM; kk += 4) {
            v2f a, bb;
            a.x  = hbuf[lo * 68 + kk + khi];
            a.y  = hbuf[lo * 68 + kk + khi + 1];
            bb.x = w2b[(kk + khi)     * DDIM + col];
            bb.y = w2b[(kk + khi + 1) * DDIM + col];
            acc = __builtin_amdgcn_wmma_f32_16x16x4_f32(
                false, a, false, bb, (short)0, acc, false, false);
        }
        const float bias = b2b[col];
#pragma unroll
        for (int v = 0; v < 8; ++v) {
            int m = v + (hi << 3);
            int t = tok_s[m];
            if (t >= 0) p.out[t * DDIM + col] = acc[v] + bias;  // coalesced
        }
    }
}

extern "C" void kernel_launch(void* const* d_in, const int* in_sizes, int n_in,
                              void* d_out, int out_size, void* d_ws, size_t ws_size,
                              hipStream_t stream) {
    (void)in_sizes; (void)n_in; (void)out_size; (void)ws_size;

    const int* ids = (const int*)d_in[0];
    int* counts = (int*)d_ws;                       // 4 ints (zeroed every call)
    int* perm   = (int*)((char*)d_ws + 256);        // 4 * NTOK ints (~1.3 MB)

    Params p;
    p.ids = ids;
    for (int b = 0; b < 4; ++b) {                   // dict order: emb,w1,b1,w2,b2
        p.emb[b] = (const float*)d_in[1 + 5 * b];
        p.w1[b]  = (const float*)d_in[2 + 5 * b];
        p.b1[b]  = (const float*)d_in[3 + 5 * b];
        p.w2[b]  = (const float*)d_in[4 + 5 * b];
        p.b2[b]  = (const float*)d_in[5 + 5 * b];
    }
    p.out    = (float*)d_out;
    p.counts = counts;
    p.perm   = perm;

    zero_counts_kernel<<<1, 32, 0, stream>>>(counts);
    bucketize_kernel<<<NTOK / 256, 256, 0, stream>>>(ids, counts, perm);
    mlp_gemm_kernel<<<dim3(NTOK / 16, 4), 128, 0, stream>>>(p);
}